// EpochedFutureFill_36206574305923
// MI455X (gfx1250) — compile-verified
//
#include <hip/hip_runtime.h>
#include <hip/hip_bf16.h>

// ---------------------------------------------------------------------------
// EpochedFutureFill == causal FIR conv (B=256, T=L=65536) via FFT convolution.
//   n_fft = 131072 = 16^4 * 2
//   forward FFT = 4 x (radix-16 Stockham stage, V_WMMA_F32_16X16X4_F32)
//                 + 1 radix-2 stage
//   inverse FFT via conjugation trick (re-uses the forward kernels)
// HBM/L2-bound on MI455X (23.3 TB/s HBM, 192MB L2); fp32 WMMA does the
// 16-point DFT butterfly math. Tile staging uses the gfx1250 async
// global->LDS DMA path (ASYNCcnt) instead of global->VGPR->ds_store.
// ---------------------------------------------------------------------------

typedef __attribute__((ext_vector_type(2))) float v2f;
typedef __attribute__((ext_vector_type(8))) float v8f;

#define NFFT   131072              // 2^17
#define SEQ_T  65536               // == NFFT/2
#define PI2    6.283185307179586f

// ---------------------------------------------------------------------------
// pad real input row (len valid elements) into complex (re, 0), zero-padded
// ---------------------------------------------------------------------------
__global__ __launch_bounds__(256)
void k_pad_complex(const float* __restrict__ src, float2* __restrict__ dst, int len)
{
    size_t tid = (size_t)blockIdx.x * 256 + threadIdx.x;   // over nbatch*NFFT
    size_t b   = tid / NFFT;
    int    i   = (int)(tid % NFFT);
    float  v   = (i < len) ? src[b * (size_t)len + i] : 0.0f;
    dst[tid] = make_float2(v, 0.0f);
}

// ---------------------------------------------------------------------------
// Radix-16 Stockham stage using fp32 WMMA.
// One wave handles one tile of 16 butterflies (a 16x16 complex matrix).
//   Y = W16 * X   with W16[m][k] = cos(th) - i sin(th), th = 2*pi*m*k/16
//   Yre = C*Xre + S*Xim ; Yim = C*Xim - S*Xre        (C=cos, S=sin)
// then twiddle by exp(-2*pi*i * p*m / (16*LS)) and scatter:
//   dst[q*16*LS + m*LS + p],  j = q*LS + p
// LS is a template constant: twiddle scale folds to a multiply, and the
// LS==1 stage skips twiddles entirely.
// ---------------------------------------------------------------------------
template <int LS, int LOG2LS>
__global__ __launch_bounds__(256)
void k_fft_radix16_wmma(const float2* __restrict__ src, float2* __restrict__ dst)
{
    // interleaved (re,im) tile per wave; col padding (17) dodges bank conflicts
    __shared__ float2 s_xy[8][16][17];

    const int lane = threadIdx.x & 31;
    const int wv   = threadIdx.x >> 5;                 // wave in block (8 waves)

    const int tiles_per_batch = (NFFT / 16) / 16;      // 512
    const long long tileg = (long long)blockIdx.x * 8 + wv;
    const int batch = (int)(tileg / tiles_per_batch);
    const int tile  = (int)(tileg % tiles_per_batch);
    const int jbase = tile * 16;                       // first butterfly id
    const size_t boff = (size_t)batch * NFFT;

    // Stage the 16x16 complex tile into LDS via the gfx1250 async DMA path
    // (GLOBAL_LOAD_ASYNC_TO_LDS_B64, tracked by ASYNCcnt): no VGPR roundtrip.
    // Consecutive lanes take consecutive cols -> contiguous 128B segments.
    for (int e = lane; e < 256; e += 32) {
        int col = e & 15, row = e >> 4;
        const float2* gp =
            src + boff + (size_t)(jbase + col) + (size_t)row * (NFFT / 16);
        // low 32 bits of the flat shared-pointer == LDS byte offset
        unsigned ldsa = (unsigned)(size_t)(&s_xy[wv][row][col]);
        asm volatile("global_load_async_to_lds_b64 %0, %1, off"
                     :: "v"(ldsa), "v"(gp) : "memory");
    }
    asm volatile("s_wait_asynccnt 0x0" ::: "memory");
    __syncthreads();

    const int half = lane >> 4;
    const int mn   = lane & 15;     // M for A fragments, N for B/C/D fragments

    v8f yre = {};
    v8f yim = {};

#pragma unroll
    for (int k = 0; k < 4; ++k) {
        // A fragments: W16 cos / sin chunks.  ISA fp32 A layout:
        //   lanes 0-15: M=lane, K = 4k + {0,1};  lanes 16-31: K = 4k + {2,3}
        v2f ac, asn, ansn;
        // B fragments from LDS (row = K, col = N), mirrored K striping.
        v2f bre, bim;
#pragma unroll
        for (int v = 0; v < 2; ++v) {
            int K = 4 * k + 2 * half + v;
            float th = (PI2 / 16.0f) * (float)(mn * K);
            float s, c;
            __sincosf(th, &s, &c);
            ac[v]   = c;
            asn[v]  = s;
            ansn[v] = -s;
            float2 t = s_xy[wv][K][mn];
            bre[v] = t.x;
            bim[v] = t.y;
        }
        // Yre += C*Xre ; Yre += S*Xim ; Yim += C*Xim ; Yim += (-S)*Xre
        yre = __builtin_amdgcn_wmma_f32_16x16x4_f32(false, ac,   false, bre,
                                                    (short)0, yre, false, false);
        yre = __builtin_amdgcn_wmma_f32_16x16x4_f32(false, asn,  false, bim,
                                                    (short)0, yre, false, false);
        yim = __builtin_amdgcn_wmma_f32_16x16x4_f32(false, ac,   false, bim,
                                                    (short)0, yim, false, false);
        yim = __builtin_amdgcn_wmma_f32_16x16x4_f32(false, ansn, false, bre,
                                                    (short)0, yim, false, false);
    }

    // Twiddle + Stockham scatter. Accumulator layout: VGPR r holds
    // M = r (lanes 0-15) or M = r+8 (lanes 16-31); N = lane&15.
    const int j = jbase + mn;
    const int p = j & (LS - 1);
    const int q = j >> LOG2LS;
#pragma unroll
    for (int r = 0; r < 8; ++r) {
        int M = r + 8 * half;
        float ore, oim;
        if constexpr (LS == 1) {
            ore = yre[r];
            oim = yim[r];
        } else {
            const float kf = -PI2 / (16.0f * (float)LS);   // compile-time
            float w = kf * (float)(p * M);
            float si, co;
            __sincosf(w, &si, &co);
            ore = yre[r] * co - yim[r] * si;
            oim = yim[r] * co + yre[r] * si;
        }
        dst[boff + (size_t)q * (size_t)(LS * 16) + (size_t)M * LS + p] =
            make_float2(ore, oim);
    }
}

// ---------------------------------------------------------------------------
// Final radix-2 stage (plain): b[j] = a[j] + w^j a[j+n/2], b[j+n/2] = diff
// (used only for the one-time filter spectrum)
// ---------------------------------------------------------------------------
__global__ __launch_bounds__(256)
void k_fft_radix2_final(const float2* __restrict__ src, float2* __restrict__ dst)
{
    size_t tid = (size_t)blockIdx.x * 256 + threadIdx.x;   // over nbatch*NFFT/2
    int    j   = (int)(tid % (NFFT / 2));
    size_t b   = tid / (NFFT / 2);
    size_t base = b * (size_t)NFFT;

    float2 a0 = src[base + j];
    float2 a1 = src[base + j + NFFT / 2];
    float ang = -PI2 * (float)j / (float)NFFT;
    float s, c;
    __sincosf(ang, &s, &c);                 // w = c + i s = exp(-2*pi*i*j/n)
    float tre = a1.x * c - a1.y * s;
    float tim = a1.x * s + a1.y * c;
    dst[base + j]            = make_float2(a0.x + tre, a0.y + tim);
    dst[base + j + NFFT / 2] = make_float2(a0.x - tre, a0.y - tim);
}

// ---------------------------------------------------------------------------
// Fused: final radix-2 of the forward FFT + Z = conj(X * F)
// ---------------------------------------------------------------------------
__global__ __launch_bounds__(256)
void k_fft_radix2_pointwise(const float2* __restrict__ src,
                            const float2* __restrict__ F,
                            float2* __restrict__ dst)
{
    size_t tid = (size_t)blockIdx.x * 256 + threadIdx.x;   // over nbatch*NFFT/2
    int    j   = (int)(tid % (NFFT / 2));
    size_t b   = tid / (NFFT / 2);
    size_t base = b * (size_t)NFFT;

    float2 a0 = src[base + j];
    float2 a1 = src[base + j + NFFT / 2];
    float ang = -PI2 * (float)j / (float)NFFT;
    float s, c;
    __sincosf(ang, &s, &c);
    float tre = a1.x * c - a1.y * s;
    float tim = a1.x * s + a1.y * c;
    float xlr = a0.x + tre, xli = a0.y + tim;     // X[j]
    float xhr = a0.x - tre, xhi = a0.y - tim;     // X[j+n/2]

    float2 fl = F[j];
    float2 fh = F[j + NFFT / 2];
    dst[base + j] =
        make_float2(xlr * fl.x - xli * fl.y, -(xlr * fl.y + xli * fl.x));
    dst[base + j + NFFT / 2] =
        make_float2(xhr * fh.x - xhi * fh.y, -(xhr * fh.y + xhi * fh.x));
}

// ---------------------------------------------------------------------------
// Fused: final radix-2 of the inverse-direction FFT + real-part writeout.
// Since SEQ_T == NFFT/2, only the lower-half butterfly output is needed:
//   y[b,j] = Re(a0 + w^j a1) * scale    for j in [0, NFFT/2)
// ---------------------------------------------------------------------------
__global__ __launch_bounds__(256)
void k_fft_radix2_writeout(const float2* __restrict__ src,
                           float* __restrict__ out, float scale)
{
    size_t tid = (size_t)blockIdx.x * 256 + threadIdx.x;   // over nbatch*NFFT/2
    int    j   = (int)(tid % (NFFT / 2));
    size_t b   = tid / (NFFT / 2);
    size_t base = b * (size_t)NFFT;

    float2 a0 = src[base + j];
    float2 a1 = src[base + j + NFFT / 2];
    float ang = -PI2 * (float)j / (float)NFFT;
    float s, c;
    __sincosf(ang, &s, &c);
    float tre = a1.x * c - a1.y * s;
    out[b * (size_t)SEQ_T + j] = (a0.x + tre) * scale;
}

// ---------------------------------------------------------------------------
extern "C" void kernel_launch(void* const* d_in, const int* in_sizes, int n_in,
                              void* d_out, int out_size, void* d_ws, size_t ws_size,
                              hipStream_t stream)
{
    const float* x    = (const float*)d_in[0];   // (256, 65536)
    const float* filt = (const float*)d_in[1];   // (1, 65536)
    float*       out  = (float*)d_out;           // (256, 65536)

    const int B   = 256;
    const int L   = in_sizes[1];                 // 65536
    const int CB  = 32;                          // batch rows per chunk
    const int NCH = B / CB;                      // 8 chunks

    // workspace layout (float2): [ Fspec | ping A | pong B ]  (~65 MB total)
    float2* Fsp = (float2*)d_ws;                          // NFFT complex
    float2* A   = Fsp + NFFT;                             // CB*NFFT complex
    float2* Bb  = A + (size_t)CB * NFFT;                  // CB*NFFT complex

    // 4 radix-16 WMMA stages; a -> b -> a -> b -> a (result back in `a`)
    auto fwd4 = [&](float2* a, float2* b, int nb) {
        k_fft_radix16_wmma<1,    0 ><<<nb * 64, 256, 0, stream>>>(a, b);
        k_fft_radix16_wmma<16,   4 ><<<nb * 64, 256, 0, stream>>>(b, a);
        k_fft_radix16_wmma<256,  8 ><<<nb * 64, 256, 0, stream>>>(a, b);
        k_fft_radix16_wmma<4096, 12><<<nb * 64, 256, 0, stream>>>(b, a);
    };

    // --- filter spectrum (one-time) -> Fsp ---------------------------------
    k_pad_complex<<<NFFT / 256, 256, 0, stream>>>(filt, A, L);
    fwd4(A, Bb, 1);
    k_fft_radix2_final<<<NFFT / 512, 256, 0, stream>>>(A, Fsp);

    // --- batched signal chunks --------------------------------------------
    const float inv_n = 1.0f / (float)NFFT;
    for (int c = 0; c < NCH; ++c) {
        const float* xc = x + (size_t)c * CB * SEQ_T;
        float*       yc = out + (size_t)c * CB * SEQ_T;

        // forward FFT of x chunk (radix-16 part) -> A
        k_pad_complex<<<CB * (NFFT / 256), 256, 0, stream>>>(xc, A, SEQ_T);
        fwd4(A, Bb, CB);

        // fused: final radix-2 + Z = conj(X*F) -> Bb
        k_fft_radix2_pointwise<<<CB * (NFFT / 512), 256, 0, stream>>>(A, Fsp, Bb);

        // inverse via forward FFT of conjugate (radix-16 part) -> Bb
        fwd4(Bb, A, CB);

        // fused: final radix-2 + y = Re(.)/n (only lower half needed)
        k_fft_radix2_writeout<<<CB * (NFFT / 512), 256, 0, stream>>>(Bb, yc, inv_n);
    }
}